// seq2seq_atten_84533546320489
// MI455X (gfx1250) — compile-verified
//
#include <hip/hip_runtime.h>
#include <math.h>

typedef __attribute__((ext_vector_type(2))) float v2f;
typedef __attribute__((ext_vector_type(8))) float v8f;

// ---------------------------------------------------------------------------
// Generic wave-level f32 WMMA GEMM:  D = A @ W^T (+Cin) (+bias) (optional relu)
//   A:  [M x K], row-major, stride lda
//   W:  [N x K], row-major, stride ldw  (B-matrix element [k][n] = W[n][k])
//   Cin/D: [M x N] with strides ldc/ldd
// Each wave computes a 32x64 tile: 2 m-halves x 4 n-subtiles (8 x v8f accs).
// B tiles are loaded once per n-subtile and feed two WMMAs (both m-halves),
// so the inner loop issues 6 loads : 8 WMMAs -> matrix-pipe bound.
// Requires M%32==0, N%64==0, K%4==0 (all shapes here satisfy this).
// Tile order is m-fastest so concurrently-resident waves share W columns
// (L0/L2 reuse while streaming W_out once from HBM).
// ---------------------------------------------------------------------------
struct GemmP {
  const float* A; int lda;
  const float* W; int ldw;
  const float* Cin; int ldc;
  const float* bias;
  float* D; int ldd;
  int M, N, K;
  int relu;
};

__device__ __forceinline__ v2f ld2(const float* p) {
  return *reinterpret_cast<const v2f*>(p);
}

__global__ __launch_bounds__(128) void gemm_wave(GemmP p0, GemmP p1) {
  GemmP p = (blockIdx.z == 0) ? p0 : p1;
  const int lane = threadIdx.x & 31;
  const int lr = lane & 15;     // row/col within 16
  const int lh = lane >> 4;     // lane half
  const int tiles_m = p.M >> 5; // 32-row supertiles
  const int tiles_n = p.N >> 6;
  const int ntiles = tiles_m * tiles_n;
  int wave = blockIdx.x * (blockDim.x >> 5) + (threadIdx.x >> 5);
  int wstride = gridDim.x * (blockDim.x >> 5);
  for (int t = wave; t < ntiles; t += wstride) {
    const int mt = t % tiles_m;          // m fastest -> concurrent waves share W tile
    const int nt = t / tiles_m;
    const int m0 = mt << 5;
    const int n0 = nt << 6;
    v8f acc[2][4] = {{{}, {}, {}, {}}, {{}, {}, {}, {}}};
    const float* arow0 = p.A + (size_t)(m0 + lr) * p.lda + 2 * lh;
    const float* arow1 = arow0 + (size_t)16 * p.lda;
    const float* wrow = p.W + (size_t)(n0 + lr) * p.ldw + 2 * lh;
    const int ldw16 = 16 * p.ldw;
    for (int k = 0; k < p.K; k += 4) {
      v2f a0 = ld2(arow0 + k);
      v2f a1 = ld2(arow1 + k);
#pragma unroll
      for (int s = 0; s < 4; ++s) {
        v2f b = ld2(wrow + s * ldw16 + k);
        acc[0][s] = __builtin_amdgcn_wmma_f32_16x16x4_f32(
            false, a0, false, b, (short)0, acc[0][s], false, false);
        acc[1][s] = __builtin_amdgcn_wmma_f32_16x16x4_f32(
            false, a1, false, b, (short)0, acc[1][s], false, false);
      }
    }
#pragma unroll
    for (int half = 0; half < 2; ++half) {
#pragma unroll
      for (int s = 0; s < 4; ++s) {
        int n = n0 + s * 16 + lr;
        float bv = p.bias ? p.bias[n] : 0.0f;
#pragma unroll
        for (int j = 0; j < 8; ++j) {
          int m = m0 + half * 16 + j + 8 * lh;
          float v = acc[half][s][j] + bv;
          if (p.Cin) v += p.Cin[(size_t)m * p.ldc + n];
          if (p.relu) v = fmaxf(v, 0.0f);
          p.D[(size_t)m * p.ldd + n] = v;
        }
      }
    }
  }
}

// ---------------------------------------------------------------------------
// Embedding gathers (float4 rows)
// ---------------------------------------------------------------------------
__global__ void embed_enc_k(const int* __restrict__ txt,
                            const float* __restrict__ emb, float* __restrict__ out) {
  int q = blockIdx.x * blockDim.x + threadIdx.x;   // B*Te*(E/4) = 2048*64
  if (q >= 2048 * 64) return;
  int row = q >> 6, c = q & 63;
  int tok = txt[row];
  reinterpret_cast<float4*>(out)[(size_t)row * 64 + c] =
      reinterpret_cast<const float4*>(emb)[(size_t)tok * 64 + c];
}

__global__ void embed_dec_k(const int* __restrict__ txt,
                            const float* __restrict__ emb, float* __restrict__ out) {
  int q = blockIdx.x * blockDim.x + threadIdx.x;   // (B*31)*(E/4) = 992*64
  if (q >= 992 * 64) return;
  int row = q >> 6, c = q & 63;
  int b = row / 31, t = row % 31;
  int tok = txt[b * 32 + t];                       // dec_texts[:, :-1]
  reinterpret_cast<float4*>(out)[(size_t)row * 64 + c] =
      reinterpret_cast<const float4*>(emb)[(size_t)tok * 64 + c];
}

// ---------------------------------------------------------------------------
// LSTM cell pointwise updates.  gate split order: i, f, g, o
// ---------------------------------------------------------------------------
__device__ __forceinline__ float sigm(float x) { return 1.0f / (1.0f + expf(-x)); }

__global__ void enc_cell_k(const float* __restrict__ g_f, const float* __restrict__ g_b,
                           float* h_f, float* c_f, float* h_b, float* c_b,
                           float* out_f, float* out_b,
                           const float* __restrict__ mask, int t) {
  int idx = blockIdx.x * blockDim.x + threadIdx.x;
  const int BH = 32 * 512;
  if (idx >= 2 * BH) return;
  int dir = idx / BH;
  int r = idx - dir * BH;
  int b = r >> 9, j = r & 511;
  const float* g = dir ? g_b : g_f;
  float* hp = dir ? h_b : h_f;
  float* cp = dir ? c_b : c_f;
  float* op = dir ? out_b : out_f;
  int tt = dir ? (63 - t) : t;
  size_t gb = (size_t)b * 2048;
  float gi = g[gb + j], gf = g[gb + 512 + j], gg = g[gb + 1024 + j], go = g[gb + 1536 + j];
  float c = cp[r], h = hp[r];
  float cn = sigm(gf) * c + sigm(gi) * tanhf(gg);
  float hn = sigm(go) * tanhf(cn);
  float m = mask[b * 64 + tt];
  h = m * hn + (1.0f - m) * h;
  c = m * cn + (1.0f - m) * c;
  hp[r] = h;
  cp[r] = c;
  op[(size_t)(b * 64 + tt) * 512 + j] = h;
}

__global__ void dec_cell_k(const float* __restrict__ g, float* h, float* c,
                           float* dec_cat, int t) {
  int idx = blockIdx.x * blockDim.x + threadIdx.x;
  if (idx >= 32 * 512) return;
  int b = idx >> 9, j = idx & 511;
  size_t gb = (size_t)b * 2048;
  float gi = g[gb + j], gf = g[gb + 512 + j], gg = g[gb + 1024 + j], go = g[gb + 1536 + j];
  float cn = sigm(gf) * c[idx] + sigm(gi) * tanhf(gg);
  float hn = sigm(go) * tanhf(cn);
  h[idx] = hn;
  c[idx] = cn;
  dec_cat[(size_t)(b * 31 + t) * 1536 + j] = hn;   // [h ; ctx] row for logits GEMM
}

// enc_output = concat(out_f, out_b) * mask
__global__ void enc_concat_k(const float* __restrict__ out_f, const float* __restrict__ out_b,
                             const float* __restrict__ mask, float* __restrict__ enc_out) {
  int idx = blockIdx.x * blockDim.x + threadIdx.x;   // 32*64*512
  if (idx >= 32 * 64 * 512) return;
  int row = idx >> 9, j = idx & 511;
  float m = mask[row];
  enc_out[(size_t)row * 1024 + j] = out_f[idx] * m;
  enc_out[(size_t)row * 1024 + 512 + j] = out_b[idx] * m;
}

// ---------------------------------------------------------------------------
// Attention: scores = df . enc_feat, softmax*mask renorm, ctx = a . enc_output
// One block per batch element (32 blocks x 256 threads).
// ---------------------------------------------------------------------------
__global__ __launch_bounds__(256) void attention_k(
    const float* __restrict__ df, const float* __restrict__ enc_feat,
    const float* __restrict__ enc_out, const float* __restrict__ mask,
    float* __restrict__ ctx, float* __restrict__ dec_cat, int t_step) {
  __shared__ float sdf[1024];
  __shared__ float ssc[64];
  __shared__ float sa[64];
  __shared__ float red[256];
  int b = blockIdx.x;
  int tid = threadIdx.x;
  for (int i = tid; i < 1024; i += 256) sdf[i] = df[(size_t)b * 1024 + i];
  __syncthreads();
  // scores: 4 threads per time step, each over 256 elems
  int tt = tid >> 2, part = tid & 3;
  const float* ef = enc_feat + (size_t)(b * 64 + tt) * 1024 + part * 256;
  float ps = 0.0f;
  for (int k = 0; k < 256; ++k) ps += sdf[part * 256 + k] * ef[k];
  red[tid] = ps;
  __syncthreads();
  if (part == 0) ssc[tt] = red[tid] + red[tid + 1] + red[tid + 2] + red[tid + 3];
  __syncthreads();
  float mx = -3.0e38f;
  for (int i = 0; i < 64; ++i) mx = fmaxf(mx, ssc[i]);
  if (tid < 64) sa[tid] = expf(ssc[tid] - mx) * mask[b * 64 + tid];
  __syncthreads();
  float sum = 0.0f;
  for (int i = 0; i < 64; ++i) sum += sa[i];
  float inv = 1.0f / sum;
  // ctx: each thread owns 4 consecutive channels
  float a0 = 0.f, a1 = 0.f, a2 = 0.f, a3 = 0.f;
  const float4* eo = reinterpret_cast<const float4*>(enc_out + (size_t)b * 64 * 1024);
  for (int i = 0; i < 64; ++i) {
    float a = sa[i] * inv;
    float4 e = eo[(size_t)i * 256 + tid];
    a0 += a * e.x; a1 += a * e.y; a2 += a * e.z; a3 += a * e.w;
  }
  float4 r = make_float4(a0, a1, a2, a3);
  reinterpret_cast<float4*>(ctx + (size_t)b * 1024)[tid] = r;
  reinterpret_cast<float4*>(dec_cat + (size_t)(b * 31 + t_step) * 1536 + 512)[tid] = r;
}

// ---------------------------------------------------------------------------
// Host side
// ---------------------------------------------------------------------------
static inline void run_gemm(const GemmP& p, hipStream_t s) {
  int tiles = (p.M / 32) * (p.N / 64);
  int blocks = (tiles + 3) / 4;
  if (blocks > 8192) blocks = 8192;
  gemm_wave<<<dim3(blocks, 1, 1), dim3(128, 1, 1), 0, s>>>(p, p);
}
static inline void run_gemm2(const GemmP& a, const GemmP& b, hipStream_t s) {
  int t0 = (a.M / 32) * (a.N / 64), t1 = (b.M / 32) * (b.N / 64);
  int tiles = t0 > t1 ? t0 : t1;
  int blocks = (tiles + 3) / 4;
  if (blocks > 8192) blocks = 8192;
  gemm_wave<<<dim3(blocks, 1, 2), dim3(128, 1, 1), 0, s>>>(a, b);
}

extern "C" void kernel_launch(void* const* d_in, const int* in_sizes, int n_in,
                              void* d_out, int out_size, void* d_ws, size_t ws_size,
                              hipStream_t stream) {
  const int*   enc_texts = (const int*)d_in[0];
  const float* enc_mask  = (const float*)d_in[1];
  const int*   dec_texts = (const int*)d_in[2];
  const float* enc_emb   = (const float*)d_in[3];
  const float* dec_emb   = (const float*)d_in[4];
  const float* Wih_f = (const float*)d_in[5];
  const float* Whh_f = (const float*)d_in[6];
  const float* b_f   = (const float*)d_in[7];
  const float* Wih_b = (const float*)d_in[8];
  const float* Whh_b = (const float*)d_in[9];
  const float* b_b   = (const float*)d_in[10];
  const float* Wih_d = (const float*)d_in[11];
  const float* Whh_d = (const float*)d_in[12];
  const float* b_d   = (const float*)d_in[13];
  const float* W_h   = (const float*)d_in[14];
  const float* b_h   = (const float*)d_in[15];
  const float* W_c   = (const float*)d_in[16];
  const float* b_c   = (const float*)d_in[17];
  const float* W_df  = (const float*)d_in[18];
  const float* b_df  = (const float*)d_in[19];
  const float* W_ef  = (const float*)d_in[20];
  const float* b_ef  = (const float*)d_in[21];
  const float* W_out = (const float*)d_in[22];
  const float* b_out = (const float*)d_in[23];
  float* out = (float*)d_out;

  // Workspace carve-up (floats). Total ~19.4M floats (~78 MB).
  float* ws = (float*)d_ws;
  size_t off = 0;
  auto alloc = [&](size_t n) { float* p = ws + off; off += n; return p; };
  float* emb     = alloc((size_t)2048 * 256);   // encoder embeddings
  float* Xf      = alloc((size_t)2048 * 2048);  // emb @ Wih_f^T + b_f
  float* Xb      = alloc((size_t)2048 * 2048);
  float* out_f   = alloc((size_t)2048 * 512);
  float* out_b   = alloc((size_t)2048 * 512);
  float* enc_out = alloc((size_t)2048 * 1024);
  float* enc_ft  = alloc((size_t)2048 * 1024);
  float* h_f = alloc(32 * 512); float* c_f = alloc(32 * 512);
  float* h_b = alloc(32 * 512); float* c_b = alloc(32 * 512);
  float* g_f = alloc((size_t)32 * 2048);
  float* g_b = alloc((size_t)32 * 2048);
  float* dec_in = alloc((size_t)992 * 256);
  float* Xd     = alloc((size_t)992 * 2048);    // dec_in @ Wih_d[:, :E]^T + b_d
  float* h_d = alloc(32 * 512); float* c_d = alloc(32 * 512);
  float* ctx = alloc((size_t)32 * 1024);
  float* g_d = alloc((size_t)32 * 2048);
  float* dfb = alloc((size_t)32 * 1024);
  float* dec_cat = alloc((size_t)992 * 1536);   // [h_t ; ctx_t] rows (b*31+t)

  // --- zero-init recurrent state + ctx0 (graph-capturable memset nodes)
  hipMemsetAsync(h_f, 0, (size_t)4 * 32 * 512 * sizeof(float), stream); // h_f,c_f,h_b,c_b
  hipMemsetAsync(ctx, 0, (size_t)32 * 1024 * sizeof(float), stream);

  // --- embeddings
  embed_enc_k<<<(2048 * 64 + 255) / 256, 256, 0, stream>>>(enc_texts, enc_emb, emb);
  embed_dec_k<<<(992 * 64 + 255) / 256, 256, 0, stream>>>(dec_texts, dec_emb, dec_in);

  // --- hoisted input projections (big GEMMs, M = whole sequence)
  run_gemm({emb, 256, Wih_f, 256, nullptr, 0, b_f, Xf, 2048, 2048, 2048, 256, 0}, stream);
  run_gemm({emb, 256, Wih_b, 256, nullptr, 0, b_b, Xb, 2048, 2048, 2048, 256, 0}, stream);
  run_gemm({dec_in, 256, Wih_d, 1280, nullptr, 0, b_d, Xd, 2048, 992, 2048, 256, 0}, stream);

  // --- bidirectional encoder: fwd(t) and bwd(63-t) fused per launch (grid.z=2)
  for (int t = 0; t < 64; ++t) {
    GemmP pf = {h_f, 512, Whh_f, 512, Xf + (size_t)t * 2048, 64 * 2048,
                nullptr, g_f, 2048, 32, 2048, 512, 0};
    GemmP pb = {h_b, 512, Whh_b, 512, Xb + (size_t)(63 - t) * 2048, 64 * 2048,
                nullptr, g_b, 2048, 32, 2048, 512, 0};
    run_gemm2(pf, pb, stream);
    enc_cell_k<<<(2 * 32 * 512 + 255) / 256, 256, 0, stream>>>(
        g_f, g_b, h_f, c_f, h_b, c_b, out_f, out_b, enc_mask, t);
  }

  // --- enc_output concat + mask, enc_feat projection
  enc_concat_k<<<(32 * 64 * 512 + 255) / 256, 256, 0, stream>>>(out_f, out_b, enc_mask, enc_out);
  run_gemm({enc_out, 1024, W_ef, 1024, nullptr, 0, b_ef, enc_ft, 1024, 2048, 1024, 1024, 0}, stream);

  // --- decoder initial state: relu([hf;hb] @ W_h^T + b_h) via column-split pairs
  run_gemm({h_f, 512, W_h, 1024, nullptr, 0, b_h, h_d, 512, 32, 512, 512, 0}, stream);
  run_gemm({h_b, 512, W_h + 512, 1024, h_d, 512, nullptr, h_d, 512, 32, 512, 512, 1}, stream);
  run_gemm({c_f, 512, W_c, 1024, nullptr, 0, b_c, c_d, 512, 32, 512, 512, 0}, stream);
  run_gemm({c_b, 512, W_c + 512, 1024, c_d, 512, nullptr, c_d, 512, 32, 512, 512, 1}, stream);

  // --- decoder recurrence (logits hoisted out; store [h;ctx] rows)
  for (int t = 0; t < 31; ++t) {
    // gates = Xd[t] + h @ Whh_d^T + ctx @ Wih_d[:, E:]^T
    run_gemm({h_d, 512, Whh_d, 512, Xd + (size_t)t * 2048, 31 * 2048,
              nullptr, g_d, 2048, 32, 2048, 512, 0}, stream);
    run_gemm({ctx, 1024, Wih_d + 256, 1280, g_d, 2048,
              nullptr, g_d, 2048, 32, 2048, 1024, 0}, stream);
    dec_cell_k<<<(32 * 512 + 255) / 256, 256, 0, stream>>>(g_d, h_d, c_d, dec_cat, t);
    // df = [h;c] @ W_df^T + b_df  (column-split pair)
    run_gemm({h_d, 512, W_df, 1024, nullptr, 0, b_df, dfb, 1024, 32, 1024, 512, 0}, stream);
    run_gemm({c_d, 512, W_df + 512, 1024, dfb, 1024, nullptr, dfb, 1024, 32, 1024, 512, 0}, stream);
    attention_k<<<32, 256, 0, stream>>>(dfb, enc_ft, enc_out, enc_mask, ctx, dec_cat, t);
  }

  // --- single batched logits GEMM: (992 x 1536) @ W_out^T + b_out -> (992 x 32000)
  run_gemm({dec_cat, 1536, W_out, 1536, nullptr, 0, b_out, out, 32000,
            992, 32000, 1536, 0}, stream);
}